// MLMPairwiseAtomicEmbedding_56023553409079
// MI455X (gfx1250) — compile-verified
//
#include <hip/hip_runtime.h>
#include <stdint.h>
#include <math.h>

typedef __attribute__((ext_vector_type(2))) float v2f;
typedef __attribute__((ext_vector_type(8))) float v8f;

#define SCALE     0.1f
#define EPS_F     1e-8f
#define D_MAX_F   20.0f
#define FEAT_DIM  452
// freq_i = exp(2i * (-ln(10000)/64)) = exp(i * FREQ_COEF)
#define FREQ_COEF (-0.28782313662425574f)
// RBF: mu_j = j * 20/63 ; 1/sigma = 64/20
#define MU_STEP   (D_MAX_F / 63.0f)
#define INV_SIGMA (64.0f / D_MAX_F)

__constant__ float c_bbl[4][3] = {
    {-0.525f, 1.363f, 0.0f},
    { 0.0f,   0.0f,   0.0f},
    { 1.526f, 0.0f,   0.0f},
    { 2.153f,-1.062f, 0.0f}
};

// raw hardware v_sqrt_f32 (~1 ulp), avoids the correctly-rounded expansion
__device__ __forceinline__ float fast_sqrtf(float x) {
    return __builtin_amdgcn_sqrtf(x);
}

__global__ __launch_bounds__(256)
void edge_feat_kernel(const float* __restrict__ atom14,
                      const float* __restrict__ rot,
                      const float* __restrict__ trans,
                      const uint8_t* __restrict__ mask,
                      const int* __restrict__ seq,
                      const int* __restrict__ eidx,
                      float* __restrict__ out, int E)
{
    const int lane = threadIdx.x & 31;
    const int e    = blockIdx.x * 8 + (threadIdx.x >> 5);
    if (e >= E) return;                       // wave-uniform exit (E % 8 == 0 in practice)

    const int  a     = lane & 15;             // atom index this lane owns (0..15, 14/15 pad)
    const bool hi    = (lane >= 16);
    const bool valid = (a < 14);

    const int dstN = eidx[e];
    const int srcN = eidx[E + e];
    const bool nS  = mask[srcN] != 0;
    const bool nD  = mask[dstN] != 0;
    const int seqS = seq[srcN];
    const int seqD = seq[dstN];

    // scalar multiplier from the LAST edge: ~(noised[src[-1]] | noised[dst[-1]])
    const float lastFlag = (mask[eidx[E - 1]] | mask[eidx[2 * E - 1]]) ? 0.0f : 1.0f;

    // src rigid frame (uniform per wave; hits L2/L0)
    float R[9], tS[3];
#pragma unroll
    for (int i = 0; i < 9; i++) R[i] = rot[srcN * 9 + i];
#pragma unroll
    for (int i = 0; i < 3; i++) tS[i] = trans[srcN * 3 + i];

    // ---- per-atom local coords: u = R^T (ps - tS),  v = R^T (pd - tS) ----
    float u0 = 0.f, u1 = 0.f, u2v = 0.f;
    float vv0 = 0.f, vv1 = 0.f, vv2 = 0.f;
    if (valid) {
        // source node atom (noised -> backbone frame, else scaled input)
        float ps0, ps1, ps2;
        if (nS) {
            if (a < 4) {
                const float* bl = c_bbl[a];
                ps0 = R[0]*bl[0] + R[1]*bl[1] + R[2]*bl[2] + tS[0];
                ps1 = R[3]*bl[0] + R[4]*bl[1] + R[5]*bl[2] + tS[1];
                ps2 = R[6]*bl[0] + R[7]*bl[1] + R[8]*bl[2] + tS[2];
            } else { ps0 = ps1 = ps2 = 0.f; }
        } else {
            const float* p = atom14 + ((size_t)srcN * 14 + a) * 3;
            ps0 = p[0] * SCALE; ps1 = p[1] * SCALE; ps2 = p[2] * SCALE;
        }
        float d0 = ps0 - tS[0], d1 = ps1 - tS[1], d2 = ps2 - tS[2];
        u0  = R[0]*d0 + R[3]*d1 + R[6]*d2;      // R^T * d
        u1  = R[1]*d0 + R[4]*d1 + R[7]*d2;
        u2v = R[2]*d0 + R[5]*d1 + R[8]*d2;

        // destination node atom, expressed in src frame
        float pd0, pd1, pd2;
        if (nD) {
            if (a < 4) {
                float Rd[9], tD[3];
#pragma unroll
                for (int i = 0; i < 9; i++) Rd[i] = rot[dstN * 9 + i];
#pragma unroll
                for (int i = 0; i < 3; i++) tD[i] = trans[dstN * 3 + i];
                const float* bl = c_bbl[a];
                pd0 = Rd[0]*bl[0] + Rd[1]*bl[1] + Rd[2]*bl[2] + tD[0];
                pd1 = Rd[3]*bl[0] + Rd[4]*bl[1] + Rd[5]*bl[2] + tD[1];
                pd2 = Rd[6]*bl[0] + Rd[7]*bl[1] + Rd[8]*bl[2] + tD[2];
            } else { pd0 = pd1 = pd2 = 0.f; }
        } else {
            const float* p = atom14 + ((size_t)dstN * 14 + a) * 3;
            pd0 = p[0] * SCALE; pd1 = p[1] * SCALE; pd2 = p[2] * SCALE;
        }
        float e0 = pd0 - tS[0], e1 = pd1 - tS[1], e2 = pd2 - tS[2];
        vv0 = R[0]*e0 + R[3]*e1 + R[6]*e2;
        vv1 = R[1]*e0 + R[4]*e1 + R[7]*e2;
        vv2 = R[2]*e0 + R[5]*e1 + R[8]*e2;
    }
    // w = v - eps (eps added to vector difference in reference)
    const float w0  = valid ? (vv0 - EPS_F) : 0.f;
    const float w1  = valid ? (vv1 - EPS_F) : 0.f;
    const float w2c = valid ? (vv2 - EPS_F) : 0.f;

    const float uu2 = u0*u0 + u1*u1 + u2v*u2v;
    const float ww2 = w0*w0 + w1*w1 + w2c*w2c;

    // ---- WMMA fragments per ISA 32-bit 16x4 layout ----
    // lanes 0-15: {K0,K1} of row/col (lane), lanes 16-31: {K2,K3(pad=0)}
    v2f Af, Bf;
    Af.x = hi ? u2v : u0;   Af.y = hi ? 0.f : u1;
    Bf.x = hi ? w2c : w0;   Bf.y = hi ? 0.f : w1;
    v8f Cz = {0.f,0.f,0.f,0.f,0.f,0.f,0.f,0.f};
    // G[M][N] = u_M . w_N  (cross term of the pairwise distances)
    v8f G = __builtin_amdgcn_wmma_f32_16x16x4_f32(
        false, Af, false, Bf, (short)0, Cz, false, false);

    // broadcast ||u_M||^2 : lane needs rows M = r + (hi?8:0), held by lane M
    float u2b[8];
#pragma unroll
    for (int r = 0; r < 8; r++)
        u2b[r] = __shfl(uu2, r + (hi ? 8 : 0), 32);

    float* base = out + (size_t)e * FEAT_DIM;

    // ---- pair block (cols 256..451), D row M = r(+8), col N = a ----
    if (valid) {
#pragma unroll
        for (int r = 0; r < 8; r++) {
            const int M = r + (hi ? 8 : 0);
            if (M < 14) {
                float g  = G[r];
                float d2 = fmaxf(u2b[r] - 2.0f * g + ww2, 0.0f);
                __builtin_nontemporal_store(fast_sqrtf(d2) * lastFlag,
                                            base + 256 + M * 14 + a);
            }
        }
        // local-frame features: lo half writes u (src local), hi half writes v
        if (!hi) {
            const float m = nS ? 0.f : 1.f;
            __builtin_nontemporal_store(u0  * m, base + 172 + 3*a + 0);
            __builtin_nontemporal_store(u1  * m, base + 172 + 3*a + 1);
            __builtin_nontemporal_store(u2v * m, base + 172 + 3*a + 2);
        } else {
            const float m = nD ? 0.f : 1.f;
            __builtin_nontemporal_store(vv0 * m, base + 214 + 3*a + 0);
            __builtin_nontemporal_store(vv1 * m, base + 214 + 3*a + 1);
            __builtin_nontemporal_store(vv2 * m, base + 214 + 3*a + 2);
        }
    }

    // ---- scalar features (cols 0..171), strided across the wave ----
    const float* xs = atom14 + ((size_t)srcN * 14 + 1) * 3;  // CA, pre-noising
    const float* xd = atom14 + ((size_t)dstN * 14 + 1) * 3;
    const float ddx = (xd[0] - xs[0]) * SCALE;
    const float ddy = (xd[1] - xs[1]) * SCALE;
    const float ddz = (xd[2] - xs[2]) * SCALE;
    const float Dca = fast_sqrtf(ddx*ddx + ddy*ddy + ddz*ddz);
    const float dpos = (float)(dstN - srcN);

    for (int c = lane; c < 172; c += 32) {
        float val;
        if (c < 44) {
            if (c == 0)       val = nD ? 1.f : 0.f;
            else if (c == 1)  val = nS ? 1.f : 0.f;
            else if (c < 23)  val = (!nS && (c - 2)  == seqS) ? 1.f : 0.f;
            else              val = (!nD && (c - 23) == seqD) ? 1.f : 0.f;
        } else if (c < 108) {
            float t = (Dca - (float)(c - 44) * MU_STEP) * INV_SIGMA;
            val = __expf(-t * t);
        } else {
            int i = (c < 140) ? (c - 108) : (c - 140);
            float ang = dpos * __expf((float)i * FREQ_COEF);
            val = (c < 140) ? __cosf(ang) : __sinf(ang);
        }
        __builtin_nontemporal_store(val, base + c);
    }
}

extern "C" void kernel_launch(void* const* d_in, const int* in_sizes, int n_in,
                              void* d_out, int out_size, void* d_ws, size_t ws_size,
                              hipStream_t stream) {
    const float*   atom14 = (const float*)d_in[0];    // (N,14,3) f32
    const float*   rotm   = (const float*)d_in[1];    // (N,3,3)  f32
    const float*   transv = (const float*)d_in[2];    // (N,3)    f32
    const uint8_t* maskv  = (const uint8_t*)d_in[3];  // (N,)     bool (1B)
    const int*     seqv   = (const int*)d_in[4];      // (N,)     i32
    const int*     eidx   = (const int*)d_in[5];      // (2,E)    i32
    float*         outp   = (float*)d_out;

    const int E = in_sizes[5] / 2;
    dim3 block(256);
    dim3 grid((E + 7) / 8);   // one wave32 per edge, 8 waves per block
    edge_feat_kernel<<<grid, block, 0, stream>>>(atom14, rotm, transv, maskv,
                                                 seqv, eidx, outp, E);
}